// Net_1984274891283
// MI455X (gfx1250) — compile-verified
//
#include <hip/hip_runtime.h>
#include <hip/hip_bf16.h>

typedef __attribute__((ext_vector_type(2))) float v2f;
typedef __attribute__((ext_vector_type(8))) float v8f;

#define IN_CH 256
#define OUT_CH 64
#define ALPHA 0.1f
#define ONE_M_ALPHA 0.9f
#define K_ITERS 10

// ---------------- degree / norm precompute ----------------

__global__ void deg_init_kernel(float* __restrict__ deg, int n) {
    int i = blockIdx.x * blockDim.x + threadIdx.x;
    if (i < n) deg[i] = 1.0f;  // self-loop
}

__global__ void deg_count_kernel(const int* __restrict__ cols, float* __restrict__ deg, int nE) {
    int e = blockIdx.x * blockDim.x + threadIdx.x;
    if (e < nE) atomicAdd(&deg[cols[e]], 1.0f);
}

__global__ void rsqrt_kernel(float* __restrict__ dinv, int n) {
    int i = blockIdx.x * blockDim.x + threadIdx.x;
    if (i < n) dinv[i] = rsqrtf(dinv[i]);  // deg >= 1 always (self-loop)
}

__global__ void scale_kernel(const int* __restrict__ rows, const int* __restrict__ cols,
                             const float* __restrict__ dinv, float* __restrict__ scale, int nE) {
    int e = blockIdx.x * blockDim.x + threadIdx.x;
    if (e < nE) scale[e] = ONE_M_ALPHA * dinv[rows[e]] * dinv[cols[e]];
}

// ---------------- fp32 WMMA GEMM: h = relu(x @ W1 + b1) ----------------
// one wave (32 lanes) computes a 16-row strip x all 64 output cols.
// 8 waves / block -> 8 M-tiles per block. W1 (64KB) staged in LDS.

__global__ __launch_bounds__(256) void gemm_relu_kernel(
    const float* __restrict__ x, const float* __restrict__ W,
    const float* __restrict__ bias, float* __restrict__ h, int n_nodes) {

    __shared__ float sW[IN_CH * OUT_CH];  // 64 KB
    for (int i = threadIdx.x; i < IN_CH * OUT_CH; i += 256) sW[i] = W[i];
    __syncthreads();

    const int wave = threadIdx.x >> 5;   // wave32
    const int lane = threadIdx.x & 31;
    const int mt = blockIdx.x * 8 + wave;
    if (mt * 16 >= n_nodes) return;      // wave-uniform: EXEC stays all-1s for WMMA

    const int lh = lane & 15;
    const int kh = (lane >> 4) * 2;      // A/B fragment K phase: 0 or 2

    v8f acc[4] = {};
    const float* __restrict__ xrow = x + (size_t)(mt * 16 + lh) * IN_CH;

    for (int k = 0; k < IN_CH; k += 4) {
        v2f a;
        a.x = xrow[k + kh];
        a.y = xrow[k + kh + 1];
#pragma unroll
        for (int nt = 0; nt < 4; ++nt) {
            v2f b;
            b.x = sW[(k + kh) * OUT_CH + nt * 16 + lh];
            b.y = sW[(k + kh + 1) * OUT_CH + nt * 16 + lh];
            acc[nt] = __builtin_amdgcn_wmma_f32_16x16x4_f32(
                false, a, false, b, (short)0, acc[nt], false, false);
        }
    }

    // D layout: VGPR r -> lanes 0-15: M=r, N=lane; lanes 16-31: M=8+r, N=lane-16
    const int mbase = mt * 16 + (lane >> 4) * 8;
#pragma unroll
    for (int nt = 0; nt < 4; ++nt) {
        const float bv = bias[nt * 16 + lh];
#pragma unroll
        for (int r = 0; r < 8; ++r) {
            float v = acc[nt][r] + bv;
            h[(size_t)(mbase + r) * OUT_CH + nt * 16 + lh] = fmaxf(v, 0.0f);
        }
    }
}

// ---------------- propagation ----------------
// out[i] = ALPHA*h[i] + (1-ALPHA)*dinv[i]^2 * z[i]   (teleport + self-loop folded)
__global__ void init_out_kernel(const float* __restrict__ h, const float* __restrict__ zin,
                                const float* __restrict__ dinv, float* __restrict__ out, int n) {
    int t = blockIdx.x * blockDim.x + threadIdx.x;
    int i = t >> 4;
    if (i >= n) return;
    int q = (t & 15) * 4;
    float di = dinv[i];
    float sl = ONE_M_ALPHA * di * di;
    const float4 hv = *(const float4*)(h + (size_t)i * OUT_CH + q);
    const float4 zv = *(const float4*)(zin + (size_t)i * OUT_CH + q);
    float4 o;
    o.x = ALPHA * hv.x + sl * zv.x;
    o.y = ALPHA * hv.y + sl * zv.y;
    o.z = ALPHA * hv.z + sl * zv.z;
    o.w = ALPHA * hv.w + sl * zv.w;
    *(float4*)(out + (size_t)i * OUT_CH + q) = o;
}

// out[col[e]] += scale[e] * z[row[e]]   — 16 lanes x float4 per edge (coalesced 256B gather)
__global__ void prop_edges_kernel(const int* __restrict__ rows, const int* __restrict__ cols,
                                  const float* __restrict__ scale, const float* __restrict__ zin,
                                  float* __restrict__ out, int nE) {
    int t = blockIdx.x * blockDim.x + threadIdx.x;
    int e = t >> 4;
    if (e >= nE) return;
    int q = (t & 15) * 4;
    int r = rows[e];
    int c = cols[e];
    float s = scale[e];
    const float4 zv = *(const float4*)(zin + (size_t)r * OUT_CH + q);
    float* o = out + (size_t)c * OUT_CH + q;
    atomicAdd(o + 0, s * zv.x);
    atomicAdd(o + 1, s * zv.y);
    atomicAdd(o + 2, s * zv.z);
    atomicAdd(o + 3, s * zv.w);
}

// ---------------- launcher ----------------

extern "C" void kernel_launch(void* const* d_in, const int* in_sizes, int n_in,
                              void* d_out, int out_size, void* d_ws, size_t ws_size,
                              hipStream_t stream) {
    const float* x  = (const float*)d_in[0];
    const int*   ei = (const int*)d_in[1];       // [2, E] int32
    const float* W1 = (const float*)d_in[2];
    const float* b1 = (const float*)d_in[3];
    float* out = (float*)d_out;

    const int N = in_sizes[0] / IN_CH;
    const int E = in_sizes[1] / 2;
    const int* rows = ei;
    const int* cols = ei + E;

    float* ws   = (float*)d_ws;
    float* h    = ws;                               // N*64
    float* zP   = h  + (size_t)N * OUT_CH;          // N*64
    float* dinv = zP + (size_t)N * OUT_CH;          // N   (deg, then dinv in-place)
    float* scl  = dinv + N;                         // E

    // norm precompute
    deg_init_kernel<<<(N + 255) / 256, 256, 0, stream>>>(dinv, N);
    deg_count_kernel<<<(E + 255) / 256, 256, 0, stream>>>(cols, dinv, E);
    rsqrt_kernel<<<(N + 255) / 256, 256, 0, stream>>>(dinv, N);
    scale_kernel<<<(E + 255) / 256, 256, 0, stream>>>(rows, cols, dinv, scl, E);

    // h = relu(x @ W1 + b1) via fp32 WMMA
    const int mtiles = (N + 15) / 16;
    gemm_relu_kernel<<<(mtiles + 7) / 8, 256, 0, stream>>>(x, W1, b1, h, N);

    // K iterations of APPNP; odd iterations write d_out so iter 9 lands in d_out
    const int nodeThreads = N * 16;
    const int edgeThreads = E * 16;
    const float* zin = h;
    for (int k = 0; k < K_ITERS; ++k) {
        float* zout = (k & 1) ? out : zP;
        init_out_kernel<<<(nodeThreads + 255) / 256, 256, 0, stream>>>(h, zin, dinv, zout, N);
        prop_edges_kernel<<<(edgeThreads + 255) / 256, 256, 0, stream>>>(rows, cols, scl, zin, zout, E);
        zin = zout;
    }
}